// CenterLoss_6811818131842
// MI455X (gfx1250) — compile-verified
//
#include <hip/hip_runtime.h>
#include <hip/hip_bf16.h>

typedef __attribute__((ext_vector_type(2))) float v2f;
typedef __attribute__((ext_vector_type(8))) float v8f;

#define NB 8192
#define NC 8192
#define ND 256
#define CLAMP_LO 1e-12f
#define CLAMP_HI 1e12f

// ---------------------------------------------------------------------------
// Row squared norms: one wave per row (wave32), 8 rows per 256-thread block.
// ---------------------------------------------------------------------------
__global__ void norms_kernel(const float* __restrict__ x, const float* __restrict__ cen,
                             float* __restrict__ sqx, float* __restrict__ sqc) {
    int gw   = blockIdx.x * (blockDim.x >> 5) + (threadIdx.x >> 5);
    int lane = threadIdx.x & 31;
    const float* src; float* dst; int row;
    if (gw < NB) { src = x;   dst = sqx; row = gw; }
    else         { src = cen; dst = sqc; row = gw - NB; }
    const float* p = src + (size_t)row * ND;
    float s = 0.f;
#pragma unroll
    for (int k = lane; k < ND; k += 32) { float v = p[k]; s += v * v; }
    for (int m = 16; m; m >>= 1) s += __shfl_xor(s, m, 32);
    if (lane == 0) dst[row] = s;
}

__global__ void init_kernel(unsigned* __restrict__ dnegbits, unsigned* __restrict__ gmaxbits) {
    int i = blockIdx.x * blockDim.x + threadIdx.x;
    if (i < NB) dnegbits[i] = 0x7F800000u;   // +inf bits
    if (i == 0) *gmaxbits = 0u;
}

// ---------------------------------------------------------------------------
// fp32 WMMA mainloop: one wave computes a 16x64 strip (4 accumulators).
// A is 16x4 fp32 (2 VGPRs/lane), B is 4x16 fp32 (2 VGPRs/lane). Per the CDNA5
// VGPR layouts, both fragments load as: float2 from row (base + lane&15) at
// column k + 2*(lane>>4)  -- a single 8-byte global load each.
// ---------------------------------------------------------------------------
__device__ __forceinline__ void wmma_mainloop(const float* __restrict__ A,
                                              const float* __restrict__ Bm,
                                              int m0, int n0, v8f* acc) {
    const int lane = threadIdx.x & 31;
    const int half = lane >> 4;
    const int r    = lane & 15;
    const float* ap  = A  + (size_t)(m0 + r) * ND;
    const float* bp0 = Bm + (size_t)(n0 + r) * ND;
    const size_t bstep = (size_t)16 * ND;
    for (int k = 0; k < ND; k += 4) {
        const int kk = k + 2 * half;
        v2f a = *(const v2f*)(ap + kk);
#pragma unroll
        for (int t = 0; t < 4; ++t) {
            v2f b = *(const v2f*)(bp0 + (size_t)t * bstep + kk);
            acc[t] = __builtin_amdgcn_wmma_f32_16x16x4_f32(
                false, a, false, b, (short)0, acc[t], false, false);
        }
    }
}

// ---------------------------------------------------------------------------
// GEMM1 fused: per-row d_pos gather + masked row-min (never stores distmat).
// Block = 8 waves; wave w owns cols [blockIdx.y*512 + w*64, +64), rows m0..m0+15.
// ---------------------------------------------------------------------------
__global__ void gemm1_kernel(const float* __restrict__ x, const float* __restrict__ cen,
                             const int* __restrict__ labels,
                             const float* __restrict__ sqx, const float* __restrict__ sqc,
                             float* __restrict__ dpos, unsigned* __restrict__ dnegbits) {
    const int w    = threadIdx.x >> 5;
    const int lane = threadIdx.x & 31;
    const int half = lane >> 4;
    const int r    = lane & 15;
    const int m0   = blockIdx.x * 16;
    const int n0   = blockIdx.y * 512 + w * 64;

    v8f zero = {};
    v8f acc[4] = { zero, zero, zero, zero };
    wmma_mainloop(x, cen, m0, n0, acc);

    int   lm[8]; float sx[8]; float rowmin[8];
#pragma unroll
    for (int v = 0; v < 8; ++v) {
        int m = m0 + v + 8 * half;
        lm[v] = labels[m];
        sx[v] = sqx[m];
        rowmin[v] = __builtin_inff();
    }
#pragma unroll
    for (int t = 0; t < 4; ++t) {
        int col = n0 + t * 16 + r;
        float sc = sqc[col];
#pragma unroll
        for (int v = 0; v < 8; ++v) {
            float d = sx[v] + sc - 2.0f * acc[t][v];
            d = fminf(fmaxf(d, CLAMP_LO), CLAMP_HI);
            if (lm[v] == col) {                       // own-class entry
                dpos[m0 + v + 8 * half] = d;          // exactly one lane hits
                d = __builtin_inff();                 // exclude from min
            }
            rowmin[v] = fminf(rowmin[v], d);
        }
    }
    // reduce across the 16 lanes of this half (masks < 16 stay in-half)
#pragma unroll
    for (int v = 0; v < 8; ++v) {
#pragma unroll
        for (int m = 8; m; m >>= 1)
            rowmin[v] = fminf(rowmin[v], __shfl_xor(rowmin[v], m, 32));
    }
    if (r == 0) {
#pragma unroll
        for (int v = 0; v < 8; ++v)
            atomicMin(&dnegbits[m0 + v + 8 * half], __float_as_uint(rowmin[v]));
    }
}

// ---------------------------------------------------------------------------
// GEMM2: distmat3 = clamp(||ci||^2 + ||cj||^2 - 2 ci.cj), stored to d_out+2.
// ---------------------------------------------------------------------------
__global__ void gemm2_kernel(const float* __restrict__ cen, const float* __restrict__ sqc,
                             float* __restrict__ d3) {
    const int w    = threadIdx.x >> 5;
    const int lane = threadIdx.x & 31;
    const int half = lane >> 4;
    const int r    = lane & 15;
    const int m0   = blockIdx.x * 16;
    const int n0   = blockIdx.y * 512 + w * 64;

    v8f zero = {};
    v8f acc[4] = { zero, zero, zero, zero };
    wmma_mainloop(cen, cen, m0, n0, acc);

    float sm[8];
#pragma unroll
    for (int v = 0; v < 8; ++v) sm[v] = sqc[m0 + v + 8 * half];
#pragma unroll
    for (int t = 0; t < 4; ++t) {
        int col = n0 + t * 16 + r;
        float sc = sqc[col];
#pragma unroll
        for (int v = 0; v < 8; ++v) {
            int m = m0 + v + 8 * half;
            float d = sm[v] + sc - 2.0f * acc[t][v];
            d = fminf(fmaxf(d, CLAMP_LO), CLAMP_HI);
            d3[(size_t)m * NC + col] = d;
        }
    }
}

// ---------------------------------------------------------------------------
// One pass over distmat3: column sums (for the [:, min_inds] gather-mean) and
// global max (for the diagonal mask). Thread <-> column, coalesced row sweeps.
// ---------------------------------------------------------------------------
__global__ void colsum_max_kernel(const float* __restrict__ d3, float* __restrict__ colsum,
                                  unsigned* __restrict__ gmaxbits) {
    int c = blockIdx.x * blockDim.x + threadIdx.x;
    float s = 0.f, mx = 0.f;
    for (int i = 0; i < NC; ++i) {
        float v = d3[(size_t)i * NC + c];
        s += v;
        mx = fmaxf(mx, v);
    }
    colsum[c] = s;
    for (int m = 16; m; m >>= 1) mx = fmaxf(mx, __shfl_xor(mx, m, 32));
    if ((threadIdx.x & 31) == 0) atomicMax(gmaxbits, __float_as_uint(mx));
}

// ---------------------------------------------------------------------------
// Per-row argmin of distmat3 + eye*gmax, ties -> lowest index (jnp.argmin).
// One wave per row.
// ---------------------------------------------------------------------------
__global__ void argmin_kernel(const float* __restrict__ d3, const unsigned* __restrict__ gmaxbits,
                              int* __restrict__ minind) {
    int row  = blockIdx.x * (blockDim.x >> 5) + (threadIdx.x >> 5);
    int lane = threadIdx.x & 31;
    float gmax = __uint_as_float(*gmaxbits);
    const float* p = d3 + (size_t)row * NC;
    float bv = __builtin_inff(); int bi = 0x7FFFFFFF;
    for (int c = lane; c < NC; c += 32) {
        float v = p[c] + (c == row ? gmax : 0.f);
        if (v < bv) { bv = v; bi = c; }               // strict < keeps first
    }
    for (int m = 16; m; m >>= 1) {
        float ov = __shfl_xor(bv, m, 32);
        int   oi = __shfl_xor(bi, m, 32);
        if (ov < bv || (ov == bv && oi < bi)) { bv = ov; bi = oi; }
    }
    if (lane == 0) minind[row] = bi;
}

__global__ void loss_cent_kernel(const float* __restrict__ dpos,
                                 const unsigned* __restrict__ dnegbits,
                                 float* __restrict__ out0) {
    __shared__ float red[8];
    float s = 0.f;
    for (int i = threadIdx.x; i < NB; i += blockDim.x)
        s += dpos[i] / (__uint_as_float(dnegbits[i]) + 1.0f);
    for (int m = 16; m; m >>= 1) s += __shfl_xor(s, m, 32);
    if ((threadIdx.x & 31) == 0) red[threadIdx.x >> 5] = s;
    __syncthreads();
    if (threadIdx.x == 0) {
        float tot = 0.f;
        for (int i = 0; i < 8; ++i) tot += red[i];
        *out0 = tot / (float)NB;
    }
}

__global__ void loss_dis3_kernel(const float* __restrict__ colsum,
                                 const int* __restrict__ minind,
                                 float* __restrict__ out1) {
    __shared__ float red[8];
    float s = 0.f;
    for (int j = threadIdx.x; j < NC; j += blockDim.x)
        s += colsum[minind[j]];
    for (int m = 16; m; m >>= 1) s += __shfl_xor(s, m, 32);
    if ((threadIdx.x & 31) == 0) red[threadIdx.x >> 5] = s;
    __syncthreads();
    if (threadIdx.x == 0) {
        float tot = 0.f;
        for (int i = 0; i < 8; ++i) tot += red[i];
        *out1 = -(float)((double)tot / ((double)NC * (double)NC));
    }
}

// ---------------------------------------------------------------------------
extern "C" void kernel_launch(void* const* d_in, const int* in_sizes, int n_in,
                              void* d_out, int out_size, void* d_ws, size_t ws_size,
                              hipStream_t stream) {
    const float* x      = (const float*)d_in[0];
    const int*   labels = (const int*)d_in[1];
    const float* cen    = (const float*)d_in[2];

    float* out = (float*)d_out;
    float* d3  = out + 2;                       // distmat3 output (C x C)

    float*    ws       = (float*)d_ws;
    float*    sqx      = ws;                    // [NB]
    float*    sqc      = ws + NB;               // [NC]
    float*    dpos     = ws + 2 * NB;           // [NB]
    unsigned* dnegbits = (unsigned*)(ws + 3 * NB); // [NB] positive-float bits
    float*    colsum   = ws + 4 * NB;           // [NC]
    int*      minind   = (int*)(ws + 5 * NB);   // [NC]
    unsigned* gmaxbits = (unsigned*)(ws + 6 * NB); // [1]

    init_kernel<<<NB / 256, 256, 0, stream>>>(dnegbits, gmaxbits);
    norms_kernel<<<(NB + NC) / 8, 256, 0, stream>>>(x, cen, sqx, sqc);

    dim3 g1(NB / 16, NC / 512);
    gemm1_kernel<<<g1, 256, 0, stream>>>(x, cen, labels, sqx, sqc, dpos, dnegbits);

    dim3 g2(NC / 16, NC / 512);
    gemm2_kernel<<<g2, 256, 0, stream>>>(cen, sqc, d3);

    colsum_max_kernel<<<NC / 256, 256, 0, stream>>>(d3, colsum, gmaxbits);
    argmin_kernel<<<NC / 8, 256, 0, stream>>>(d3, gmaxbits, minind);
    loss_cent_kernel<<<1, 256, 0, stream>>>(dpos, dnegbits, out);
    loss_dis3_kernel<<<1, 256, 0, stream>>>(colsum, minind, out + 1);
}